// BaseModel_9552007266854
// MI455X (gfx1250) — compile-verified
//
#include <hip/hip_runtime.h>
#include <cstddef>
#include <cstdint>

typedef __attribute__((ext_vector_type(2))) float v2f;
typedef __attribute__((ext_vector_type(8))) float v8f;

static __device__ __forceinline__ float siluf(float x){ return x / (1.0f + __expf(-x)); }

// ---------------- conv2d 3x3 stride2 pad1, writes into channel-slice of concat buffer ----
__global__ void conv2d_s2_kernel(const float* __restrict__ x, const float* __restrict__ w,
                                 const float* __restrict__ bias, float* __restrict__ y,
                                 int B,int Cin,int Cout,int Hin,int Win,int CtotOut,int Coff)
{
  int Ho = Hin >> 1, Wo = Win >> 1;
  int idx = blockIdx.x*blockDim.x + threadIdx.x;
  int total = B*Cout*Ho*Wo;
  if (idx >= total) return;
  int wo = idx % Wo; int t = idx / Wo;
  int ho = t % Ho;  t /= Ho;
  int co = t % Cout; int bb = t / Cout;
  float acc = bias[co];
  int hi0 = ho*2 - 1, wi0 = wo*2 - 1;
  for (int ci = 0; ci < Cin; ++ci){
    const float* xp = x + (size_t)(bb*Cin + ci)*Hin*Win;
    const float* wp = w + (size_t)(co*Cin + ci)*9;
    #pragma unroll
    for (int kh = 0; kh < 3; ++kh){
      int hi = hi0 + kh; if ((unsigned)hi >= (unsigned)Hin) continue;
      #pragma unroll
      for (int kw = 0; kw < 3; ++kw){
        int wi = wi0 + kw; if ((unsigned)wi >= (unsigned)Win) continue;
        acc += xp[hi*Win + wi] * wp[kh*3 + kw];
      }
    }
  }
  y[(size_t)(bb*CtotOut + Coff + co)*(Ho*Wo) + ho*Wo + wo] = acc;
}

// ---------------- per-token layernorm over C; input channel-major (b,Ctot,L), output token-major (b*L, C)
__global__ void layernorm_kernel(const float* __restrict__ x, const float* __restrict__ g,
                                 const float* __restrict__ b, float* __restrict__ xn,
                                 int B,int C,int L,int Ctot)
{
  int tok = blockIdx.x*blockDim.x + threadIdx.x;
  if (tok >= B*L) return;
  int bb = tok / L, l = tok % L;
  const float* xp = x + (size_t)bb*Ctot*L + l;
  float mu = 0.f;
  for (int c = 0; c < C; ++c) mu += xp[(size_t)c*L];
  mu *= (1.0f/C);
  float var = 0.f;
  for (int c = 0; c < C; ++c){ float d = xp[(size_t)c*L] - mu; var += d*d; }
  var *= (1.0f/C);
  float inv = rsqrtf(var + 1e-5f);
  float* o = xn + (size_t)tok*C;
  for (int c = 0; c < C; ++c) o[c] = (xp[(size_t)c*L] - mu)*inv*g[c] + b[c];
}

// ---------------- fp32 WMMA GEMM: Y[M,N] = X[M,K] * W[N,K]^T (+bias) (+softplus)
// one wave per 16x16 output tile; V_WMMA_F32_16X16X4_F32 over K in steps of 4.
// IN_MODE 0: X token-major (row m = token, contiguous K)
// IN_MODE 1: X channel-major: X[((b*in_cs)+in_koff+k)*Lin + l], m = b*Lin + l
// out_mode 0: Y[m*N+n] ; out_mode 1: Y[((b*out_cs)+n)*Lout + l], m = b*Lout + l
template<int IN_MODE>
__global__ void gemm_wmma_kernel(const float* __restrict__ X, const float* __restrict__ W,
        const float* __restrict__ bias, float* __restrict__ Y,
        int M,int N,int K,
        int in_cs,int in_koff,int Lin,
        int out_mode,int out_cs,int Lout,int act)
{
  int wave = threadIdx.x >> 5;
  int lane = threadIdx.x & 31;
  int ntn  = (N + 15) >> 4;
  int ntm  = M >> 4;                       // M is always a multiple of 16 here
  int tile = blockIdx.x*(blockDim.x >> 5) + wave;
  if (tile >= ntm*ntn) return;             // wave-uniform: EXEC all-1s at every WMMA
  int mt = tile / ntn, nt = tile % ntn;
  int half = lane >> 4, lid = lane & 15;
  v8f c = {};
  // A frag (ISA 7.12.2, 16x4 f32): lanes 0-15 hold {K0,K1}, lanes 16-31 hold {K2,K3}, M = lid
  int mA = (mt << 4) + lid;
  int nB = (nt << 4) + lid;
  int nBc = (nB < N) ? nB : (N - 1);       // clamped: cols >= N feed lanes never stored
  const float* wp = W + (size_t)nBc*K + (half << 1);
  const float* xp;
  size_t xstep;
  int bbA = 0, lA = 0;
  if (IN_MODE == 0){
    xp = X + (size_t)mA*K + (half << 1);
    xstep = 4;
  } else {
    bbA = mA / Lin; lA = mA % Lin;
    xp = X + ((size_t)bbA*in_cs + in_koff + (half << 1))*Lin + lA;
    xstep = (size_t)4*Lin;
  }
  int K4 = K & ~3;
  #pragma unroll 2
  for (int k0 = 0; k0 < K4; k0 += 4){      // branchless hot loop: 2 loads + 1 WMMA
    v2f a, b;
    if (IN_MODE == 0){
      a = *(const v2f*)xp;                 // global_load_b64 (8B-aligned: K % 4 == 0 here)
    } else {
      a.x = xp[0]; a.y = xp[Lin];
    }
    b = *(const v2f*)wp;                   // global_load_b64
    c = __builtin_amdgcn_wmma_f32_16x16x4_f32(false, a, false, b, (short)0, c, false, false);
    xp += xstep; wp += 4;
  }
  if (K & 3){                              // branchless tail (only dt_rank = 1,2 GEMMs)
    int ka  = K4 + (half << 1);
    int k0c = (ka   < K) ? ka   : (K - 1);
    int k1c = (ka+1 < K) ? ka+1 : (K - 1);
    float m0 = (ka   < K) ? 1.f : 0.f;
    float m1 = (ka+1 < K) ? 1.f : 0.f;
    const float* wr = W + (size_t)nBc*K;
    v2f a, b;
    if (IN_MODE == 0){
      const float* xr = X + (size_t)mA*K;
      a.x = xr[k0c]*m0; a.y = xr[k1c]*m1;
    } else {
      const float* xr = X + ((size_t)bbA*in_cs + in_koff)*Lin + lA;
      a.x = xr[(size_t)k0c*Lin]*m0; a.y = xr[(size_t)k1c*Lin]*m1;
    }
    b.x = wr[k0c]*m0; b.y = wr[k1c]*m1;
    c = __builtin_amdgcn_wmma_f32_16x16x4_f32(false, a, false, b, (short)0, c, false, false);
  }
  // C/D layout: VGPR v -> row v (lanes 0-15) / row v+8 (lanes 16-31), col = lid
  #pragma unroll
  for (int v = 0; v < 8; ++v){
    int m = (mt << 4) + v + (half << 3);
    int n = (nt << 4) + lid;
    if (m < M && n < N){
      float val = c[v];
      if (bias) val += bias[n];
      if (act == 1) val = (val > 20.f) ? val : log1pf(__expf(val));
      if (out_mode == 0) Y[(size_t)m*N + n] = val;
      else { int bb = m / Lout, l = m % Lout; Y[((size_t)bb*out_cs + n)*Lout + l] = val; }
    }
  }
}

// ---------------- depthwise causal conv1d (k=4, left pad 3) + bias + SiLU; reads x-half of xz
__global__ void dwconv1d_silu_kernel(const float* __restrict__ xz, const float* __restrict__ w,
        const float* __restrict__ b, float* __restrict__ out, int B,int dinner,int L)
{
  int idx = blockIdx.x*blockDim.x + threadIdx.x;
  int total = B*dinner*L;
  if (idx >= total) return;
  int l = idx % L; int t = idx / L;
  int d = t % dinner; int bb = t / dinner;
  const float* xp = xz + ((size_t)(bb*2*dinner) + d)*L;
  const float* wp = w + (size_t)d*4;
  float acc = b[d];
  #pragma unroll
  for (int k = 0; k < 4; ++k){ int li = l - 3 + k; if (li >= 0) acc += wp[k]*xp[li]; }
  out[((size_t)bb*dinner + d)*L + l] = siluf(acc);
}

// ---------------- token permutation of a channel-major (B,C2,L) tensor
// mode 1: reverse (dst j <- src L-1-j); mode 2: slice (dst j <- src (j%k)*(L/k)+j/k)
__global__ void permute_kernel(const float* __restrict__ src, float* __restrict__ dst,
        int B,int C2,int L,int mode,int k)
{
  int idx = blockIdx.x*blockDim.x + threadIdx.x;
  int total = B*C2*L;
  if (idx >= total) return;
  int j = idx % L; size_t base = (size_t)(idx / L)*L;
  int s;
  if (mode == 1) s = L - 1 - j;
  else { int Lk = L / k; s = (j % k)*Lk + j / k; }
  dst[base + j] = src[base + s];
}

// ---------------- selective scan: one thread per (b,d), 16-state recurrence in registers
__global__ void scan_kernel(const float* __restrict__ u, const float* __restrict__ dt,
        const float* __restrict__ xdbl, const float* __restrict__ A_log,
        const float* __restrict__ Dp, const float* __restrict__ xz,
        float* __restrict__ y, int B,int dinner,int L,int Ktot,int dt_rank)
{
  int idx = blockIdx.x*blockDim.x + threadIdx.x;
  if (idx >= B*dinner) return;
  int bb = idx / dinner, d = idx % dinner;
  float A[16], h[16];
  #pragma unroll
  for (int n = 0; n < 16; ++n){ A[n] = -__expf(A_log[d*16 + n]); h[n] = 0.f; }
  const float* up  = u    + ((size_t)bb*dinner + d)*L;
  const float* dtp = dt   + ((size_t)bb*dinner + d)*L;
  const float* Bp  = xdbl + ((size_t)bb*Ktot + dt_rank)*L;
  const float* Cp  = xdbl + ((size_t)bb*Ktot + dt_rank + 16)*L;
  const float* zp  = xz   + ((size_t)(bb*2*dinner) + dinner + d)*L;
  float* yp = y + ((size_t)bb*dinner + d)*L;
  float Dd = Dp[d];
  for (int l = 0; l < L; ++l){
    float dtv = dtp[l], uv = up[l];
    float dbu = dtv*uv;
    float acc = 0.f;
    #pragma unroll
    for (int n = 0; n < 16; ++n){
      float hn = __expf(dtv*A[n])*h[n] + dbu*Bp[(size_t)n*L + l];
      h[n] = hn;
      acc += hn*Cp[(size_t)n*L + l];
    }
    float zv = zp[l];
    yp[l] = (acc + uv*Dd)*siluf(zv);
  }
}

// ---------------- out = yf + reverse(yb) + unslice(ys)
__global__ void combine_kernel(const float* __restrict__ yf, const float* __restrict__ yb,
        const float* __restrict__ ys, float* __restrict__ out, int B,int dinner,int L,int k)
{
  int idx = blockIdx.x*blockDim.x + threadIdx.x;
  int total = B*dinner*L;
  if (idx >= total) return;
  int l = idx % L; size_t base = (size_t)(idx / L)*L;
  int Lk = L / k;
  int js = (l % Lk)*k + l / Lk;
  out[base + l] = yf[base + l] + yb[base + (L - 1 - l)] + ys[base + js];
}

// ---------------- mean-pool(8x8) -> fc1(256x128)+ELU -> fc2(6x256)
__global__ void head_kernel(const float* __restrict__ x4, const float* __restrict__ fc1w,
        const float* __restrict__ fc1b, const float* __restrict__ fc2w,
        const float* __restrict__ fc2b, float* __restrict__ logits)
{
  __shared__ float g[128];
  __shared__ float h[256];
  int bb = blockIdx.x, t = threadIdx.x;
  for (int c = t; c < 128; c += blockDim.x){
    const float* p = x4 + ((size_t)bb*128 + c)*64;
    float s = 0.f;
    for (int i = 0; i < 64; ++i) s += p[i];
    g[c] = s*(1.0f/64.0f);
  }
  __syncthreads();
  for (int o = t; o < 256; o += blockDim.x){
    float s = fc1b[o];
    for (int c = 0; c < 128; ++c) s += fc1w[o*128 + c]*g[c];
    h[o] = (s > 0.f) ? s : (__expf(s) - 1.f);
  }
  __syncthreads();
  if (t < 6){
    float s = fc2b[t];
    for (int c = 0; c < 256; ++c) s += fc2w[t*256 + c]*h[c];
    logits[bb*6 + t] = s;
  }
}

__global__ void copy_kernel(const float* __restrict__ src, float* __restrict__ dst, int n){
  int i = blockIdx.x*blockDim.x + threadIdx.x;
  if (i < n) dst[i] = src[i];
}

// =====================================================================================
extern "C" void kernel_launch(void* const* d_in, const int* in_sizes, int n_in,
                              void* d_out, int out_size, void* d_ws, size_t ws_size,
                              hipStream_t stream)
{
  (void)in_sizes; (void)n_in; (void)out_size; (void)ws_size;
  const int THREADS = 256;
  const float* X = (const float*)d_in[0];
  auto P = [&](int i)->const float*{ return (const float*)d_in[i]; };
  // jax-pytree flatten order: x; then params dict sorted:
  //  down1.b/w, down11.b/w, down2.b/w, down22.b/w, down3.b/w, down33.b/w, down4.b/w (idx 1..14)
  //  fc1_b(15) fc1_w(16) fc2_b(17) fc2_w(18)
  //  stages[s][d] blocks, each 25 leaves:
  //   b-branch{A_log,D,conv_b,conv_w,dt_b,dt_w,x_proj_w}(0-6), f-branch(7-13),
  //   in_proj_w(14), ln_b(15), ln_g(16), out_proj_w(17), s-branch(18-24)
  auto BLp = [&](int s,int dpt,int leaf)->const float*{
    return (const float*)d_in[19 + (s*2 + dpt)*25 + leaf];
  };

  // ---- workspace carve (floats) ----
  float* ws = (float*)d_ws;
  size_t off = 0;
  auto alloc = [&](size_t n)->float*{ float* p = ws + off; off += n; return p; };
  float* cat1 = alloc(16u*32*64*64);   // stage0 io + down11
  float* cat2 = alloc(16u*64*32*32);
  float* cat3 = alloc(16u*128*16*16);
  float* x4b  = alloc(16u*128*8*8);
  float* xn   = alloc(16u*4096*16);    // max B*L*C
  float* xz   = alloc(16u*64*4096);    // max B*4C*L
  float* xzp  = alloc(16u*64*4096);    // permuted xz; reused as branch-sum
  float* ucv  = alloc(16u*32*4096);    // max B*2C*L
  float* xdbl = alloc(16u*33*4096);    // max B*(dt_rank+32)*L
  float* dtb  = alloc(16u*32*4096);
  float* yf   = alloc(16u*32*4096);
  float* yb   = alloc(16u*32*4096);
  float* ysl  = alloc(16u*32*4096);

  auto g1 = [&](long total)->dim3{ return dim3((unsigned)((total + THREADS - 1)/THREADS)); };

  auto gemm = [&](const float* Xp,const float* Wp,const float* biasp,float* Yp,
                  int M,int N,int K,int im,int ics,int ikoff,int Lin,
                  int om,int ocs,int Lout,int act){
    int tiles  = (M/16)*((N + 15)/16);
    int blocks = (tiles + 7)/8;                       // 8 waves (tiles) per 256-thread block
    if (im == 0)
      gemm_wmma_kernel<0><<<blocks, 256, 0, stream>>>(Xp, Wp, biasp, Yp, M, N, K,
                                                      ics, ikoff, Lin, om, ocs, Lout, act);
    else
      gemm_wmma_kernel<1><<<blocks, 256, 0, stream>>>(Xp, Wp, biasp, Yp, M, N, K,
                                                      ics, ikoff, Lin, om, ocs, Lout, act);
  };

  // ---- one Mamba stage (DEPTH=2 blocks) operating in-place on channels [0,C) of buf ----
  auto run_stage = [&](float* buf,int Ctot,int C,int L,int k,int s){
    int dinner = 2*C, E = 4*C;
    int dt_rank = (C + 15)/16, Ktot = dt_rank + 32;
    int M = 16*L;
    float* ybufs[3]   = { yf, yb, ysl };
    int    modes[3]   = { 0, 1, 2 };    // forward, reverse, slice
    int    lbase[3]   = { 7, 0, 18 };   // f-, b-, s-branch leaf bases
    for (int dpt = 0; dpt < 2; ++dpt){
      layernorm_kernel<<<g1((long)M), THREADS, 0, stream>>>(buf, BLp(s,dpt,16), BLp(s,dpt,15),
                                                            xn, 16, C, L, Ctot);
      // in_proj: (B*L,C) x (4C,C)^T -> xz (B,4C,L)
      gemm(xn, BLp(s,dpt,14), nullptr, xz, M, E, C, 0,0,0,L, 1,E,L, 0);
      for (int br = 0; br < 3; ++br){
        const float* A_log  = BLp(s,dpt,lbase[br]+0);
        const float* Dp     = BLp(s,dpt,lbase[br]+1);
        const float* convb  = BLp(s,dpt,lbase[br]+2);
        const float* convw  = BLp(s,dpt,lbase[br]+3);
        const float* dtbias = BLp(s,dpt,lbase[br]+4);
        const float* dtw    = BLp(s,dpt,lbase[br]+5);
        const float* xprojw = BLp(s,dpt,lbase[br]+6);
        const float* xz_use = xz;
        if (modes[br] != 0){
          permute_kernel<<<g1((long)16*E*L), THREADS, 0, stream>>>(xz, xzp, 16, E, L,
                                                                   modes[br], k);
          xz_use = xzp;
        }
        dwconv1d_silu_kernel<<<g1((long)16*dinner*L), THREADS, 0, stream>>>(
            xz_use, convw, convb, ucv, 16, dinner, L);
        // x_proj: (B*L,2C) x (Ktot,2C)^T -> xdbl (B,Ktot,L)
        gemm(ucv, xprojw, nullptr, xdbl, M, Ktot, dinner, 1,dinner,0,L, 1,Ktot,L, 0);
        // dt: (B*L,dt_rank) x (2C,dt_rank)^T + dt_b, softplus -> dtb (B,2C,L)
        gemm(xdbl, dtw, dtbias, dtb, M, dinner, dt_rank, 1,Ktot,0,L, 1,dinner,L, 1);
        scan_kernel<<<g1((long)16*dinner), THREADS, 0, stream>>>(
            ucv, dtb, xdbl, A_log, Dp, xz_use, ybufs[br], 16, dinner, L, Ktot, dt_rank);
      }
      combine_kernel<<<g1((long)16*dinner*L), THREADS, 0, stream>>>(yf, yb, ysl, xzp,
                                                                    16, dinner, L, k);
      // out_proj: (B*L,2C) x (C,2C)^T -> back into buf channels [0,C)
      gemm(xzp, BLp(s,dpt,17), nullptr, buf, M, C, dinner, 1,dinner,0,L, 1,Ctot,L, 0);
    }
  };

  // ---- forward graph ----
  conv2d_s2_kernel<<<g1(16L*16*64*64), THREADS, 0, stream>>>(X, P(2),  P(1),  cat1, 16,3,16,128,128, 32,0);
  run_stage(cat1, 32, 16, 4096, 64, 0);
  conv2d_s2_kernel<<<g1(16L*16*64*64), THREADS, 0, stream>>>(X, P(4),  P(3),  cat1, 16,3,16,128,128, 32,16);

  conv2d_s2_kernel<<<g1(16L*32*32*32), THREADS, 0, stream>>>(cat1, P(6),  P(5),  cat2, 16,32,32,64,64, 64,0);
  run_stage(cat2, 64, 32, 1024, 32, 1);
  conv2d_s2_kernel<<<g1(16L*32*32*32), THREADS, 0, stream>>>(cat1, P(8),  P(7),  cat2, 16,32,32,64,64, 64,32);

  conv2d_s2_kernel<<<g1(16L*64*16*16), THREADS, 0, stream>>>(cat2, P(10), P(9),  cat3, 16,64,64,32,32, 128,0);
  run_stage(cat3, 128, 64, 256, 16, 2);
  conv2d_s2_kernel<<<g1(16L*64*16*16), THREADS, 0, stream>>>(cat2, P(12), P(11), cat3, 16,64,64,32,32, 128,64);

  conv2d_s2_kernel<<<g1(16L*128*8*8), THREADS, 0, stream>>>(cat3, P(14), P(13), x4b, 16,128,128,16,16, 128,0);
  run_stage(x4b, 128, 128, 64, 4, 3);

  float* outp = (float*)d_out;
  head_kernel<<<16, 256, 0, stream>>>(x4b, P(16), P(15), P(18), P(17), outp);
  copy_kernel<<<g1(131072L), THREADS, 0, stream>>>(x4b, outp + 96, 131072);
}